// HBVMul_403726926486
// MI455X (gfx1250) — compile-verified
//
#include <hip/hip_runtime.h>
#include <math.h>

// ---------------- problem constants (from reference) ----------------
#define T_STEPS 730
#define NGRID   2000
#define NMUL    16
#define LENF    15
#define PRECS_F 1e-5f
#define PWID    (12 * NMUL + 2)   // 194 params per grid cell

// qbuf layout: per grid cell, [15 zero pad][730 Qsimave][23 zero tail] = 768
#define QPAD    15
#define QSTRIDE 768
#define NTILE   46                // ceil(730 / 16) time tiles for routing

__device__ __constant__ float LOB[12] = {0.f, 50.f, 0.05f, 0.01f, 0.001f, 0.2f,
                                         0.f, 0.f, -2.5f, 0.5f, 0.f, 0.f};
__device__ __constant__ float HIB[12] = {1.f, 1000.f, 0.9f, 0.5f, 0.2f, 1.f,
                                         10.f, 100.f, 2.5f, 10.f, 0.1f, 0.2f};

typedef __attribute__((ext_vector_type(2))) float v2f;
typedef __attribute__((ext_vector_type(8))) float v8f;

// ---------------- kernel 2: gamma routing weights (one thread / grid) -------
__global__ __launch_bounds__(256) void hbv_weights(const float* __restrict__ params,
                                                   float* __restrict__ wbuf) {
    int g = blockIdx.x * blockDim.x + threadIdx.x;
    if (g >= NGRID) return;
    float pa = params[(long)g * PWID + 12 * NMUL + 0];
    float pb = params[(long)g * PWID + 12 * NMUL + 1];
    float aa = fmaxf(pa * 2.9f, 0.f) + 0.1f;        // relu(tempa)+0.1
    float th = fmaxf(pb * 6.5f, 0.f) + 0.5f;        // relu(tempb)+0.5
    float lg  = lgammaf(aa);
    float lth = logf(th);
    float w[LENF];
    float s = 0.f;
#pragma unroll
    for (int k = 0; k < LENF; ++k) {
        float tk = (float)k + 0.5f;
        float wk = expf(-lg - aa * lth + (aa - 1.f) * logf(tk) - tk / th);
        w[k] = wk;
        s += wk;
    }
    float inv = 1.f / s;
#pragma unroll
    for (int k = 0; k < LENF; ++k) wbuf[g * 16 + k] = w[k] * inv;
    wbuf[g * 16 + 15] = 0.f;                        // pad tap so K=16 for WMMA
}

// ---------------- kernel 1: fused HBV scan --------------------------------
// One thread per (grid, mul): lane = 16*half + m, wave32 covers 2 grid cells.
// States live in registers for all 730 steps; mean over NMUL via shfl_xor.
__global__ __launch_bounds__(256) void hbv_scan(const float* __restrict__ x,
                                                const float* __restrict__ params,
                                                float* __restrict__ qbuf) {
    const int lane = threadIdx.x & 31;
    const int wv   = (blockIdx.x * blockDim.x + threadIdx.x) >> 5;
    const int half = lane >> 4;
    const int m    = lane & 15;
    const int g    = wv * 2 + half;                 // 1000 waves * 2 = 2000 exact

    const float* pg = params + (long)g * PWID;
    float P[12];
#pragma unroll
    for (int i = 0; i < 12; ++i)
        P[i] = pg[i * NMUL + m] * (HIB[i] - LOB[i]) + LOB[i];
    const float BETA = P[0], FC = P[1], K0 = P[2], K1 = P[3], K2 = P[4],
                LP = P[5], PERCmax = P[6], UZL = P[7], TT = P[8],
                CFMAX = P[9], CFR = P[10], CWH = P[11];
    const float invFC = 1.f / FC, invLPFC = 1.f / (LP * FC);

    float SNOWPACK = 0.001f, MELTWATER = 0.001f, SM = 0.001f,
          SUZ = 0.001f, SLZ = 0.001f;

    float* qg = qbuf + (long)g * QSTRIDE;
    if (m == 0) {                                   // zero pad regions each call
        for (int i = 0; i < QPAD; ++i) qg[i] = 0.f;
        for (int i = QPAD + T_STEPS; i < QSTRIDE; ++i) qg[i] = 0.f;
    }

    const float4* xv = (const float4*)x;            // (T, NGRID, 4) f32
    for (int t = 0; t < T_STEPS; ++t) {
        float4 xt = xv[t * NGRID + g];
        float Pt = xt.x, Tt = xt.y, PETt = xt.w;

        bool  cold = Tt < TT;
        float RAIN = cold ? 0.f : Pt;
        float SNOW = cold ? Pt : 0.f;
        SNOWPACK += SNOW;
        float melt = fminf(fmaxf(CFMAX * (Tt - TT), 0.f), SNOWPACK);
        MELTWATER += melt;  SNOWPACK -= melt;
        float refr = fminf(fmaxf(CFR * CFMAX * (TT - Tt), 0.f), MELTWATER);
        SNOWPACK += refr;   MELTWATER -= refr;
        float tosoil = fmaxf(MELTWATER - CWH * SNOWPACK, 0.f);
        MELTWATER -= tosoil;
        float sw = fminf(fmaxf(powf(SM * invFC, BETA), 0.f), 1.f);
        float recharge = (RAIN + tosoil) * sw;
        SM += RAIN + tosoil - recharge;
        float excess = fmaxf(SM - FC, 0.f);
        SM -= excess;
        float ef = fminf(fmaxf(SM * invLPFC, 0.f), 1.f);
        float ET = fminf(SM, PETt * ef);
        SM = fmaxf(SM - ET, PRECS_F);
        SUZ += recharge + excess;
        float PERC = fminf(SUZ, PERCmax);
        SUZ -= PERC;
        float Q0 = K0 * fmaxf(SUZ - UZL, 0.f);
        SUZ -= Q0;
        float Q1 = K1 * SUZ;
        SUZ -= Q1;
        SLZ += PERC;
        float Q2 = K2 * SLZ;
        SLZ -= Q2;

        // mean over the 16 NMUL lanes (xor of bits 0-3 stays in 16-lane group)
        float q = Q0 + Q1 + Q2;
        q += __shfl_xor(q, 1, 32);
        q += __shfl_xor(q, 2, 32);
        q += __shfl_xor(q, 4, 32);
        q += __shfl_xor(q, 8, 32);
        if (m == 0) qg[QPAD + t] = q * (1.f / 16.f);
    }
}

// ---------------- kernel 3: routing conv via f32 WMMA row-sum -------------
// One wave per (grid g, 16-step time tile). A[t,k] = w[k,g]*Q[t0+t-k,g] in the
// documented f32 16x4 A layout; B = all-ones -> D rows = exact f32 K-sums.
// Chain 4x K=4 WMMAs to cover K=16 (tap 15 is zero-weighted padding).
__global__ __launch_bounds__(256) void hbv_route_wmma(const float* __restrict__ qbuf,
                                                      const float* __restrict__ wbuf,
                                                      float* __restrict__ out) {
    const int lane = threadIdx.x & 31;
    const int wv   = (blockIdx.x * blockDim.x + threadIdx.x) >> 5;
    const int g    = wv / NTILE;                    // exact: 92000 waves launched
    const int t0   = (wv % NTILE) * 16;
    const int row  = lane & 15;                     // A-matrix row M
    const int klo  = (lane >> 4) * 2;               // lanes 0-15: K=0,1; 16-31: K=2,3

    const float* qg = qbuf + (long)g * QSTRIDE + QPAD + t0 + row;
    const float* wg = wbuf + g * 16;

    v2f bones = {1.f, 1.f};                         // B = ones (layout-invariant)
    v8f acc = {};
#pragma unroll
    for (int c = 0; c < 4; ++c) {
        int k0 = 4 * c + klo;
        v2f a;
        a.x = qg[-k0]       * wg[k0];               // Q[t-k]*w[k], exact f32
        a.y = qg[-(k0 + 1)] * wg[k0 + 1];
        acc = __builtin_amdgcn_wmma_f32_16x16x4_f32(false, a, false, bones,
                                                    (short)0, acc, false, false);
    }
    // D layout: reg r -> M=r (lanes 0-15) / M=r+8 (lanes 16-31); all N equal.
    if ((lane & 15) == 0) {
        int tb = t0 + (lane >> 4) * 8;
#pragma unroll
        for (int r = 0; r < 8; ++r) {
            int t = tb + r;
            if (t < T_STEPS) out[t * NGRID + g] = acc[r];
        }
    }
}

// ---------------- host side ------------------------------------------------
extern "C" void kernel_launch(void* const* d_in, const int* in_sizes, int n_in,
                              void* d_out, int out_size, void* d_ws, size_t ws_size,
                              hipStream_t stream) {
    const float* x      = (const float*)d_in[0];    // (730, 2000, 4) f32
    const float* params = (const float*)d_in[1];    // (2000, 194)   f32
    float* out  = (float*)d_out;                    // (730, 2000, 1) f32

    float* qbuf = (float*)d_ws;                           // 2000*768 f32 (~6.1 MB)
    float* wbuf = qbuf + (size_t)NGRID * QSTRIDE;         // 2000*16 f32

    hbv_weights<<<(NGRID + 255) / 256, 256, 0, stream>>>(params, wbuf);
    hbv_scan<<<(NGRID / 2) / 8, 256, 0, stream>>>(x, params, qbuf);   // 125 blocks
    hbv_route_wmma<<<(NGRID * NTILE) / 8, 256, 0, stream>>>(qbuf, wbuf, out); // 11500 blocks
}